// clDiceLoss_21698174779876
// MI455X (gfx1250) — compile-verified
//
#include <hip/hip_runtime.h>
#include <hip/hip_bf16.h>
#include <math.h>

// Problem constants (from reference: B=16, C=1, H=W=1024)
#define HH    1024
#define WW    1024
#define BATCH 16
#define NPIX  (HH * WW)          // 1,048,576
#define NTOT  (BATCH * NPIX)     // 16,777,216
#define ITERS 5
#define EPSF  1e-6f

// Fused-iteration tile: 64x16 output tile, halo 2 for x, halo 1 for eroded.
#define TX  64
#define TY  16
#define HXX (TX + 4)   // 68
#define HYY (TY + 4)   // 20
#define EXX (TX + 2)   // 66
#define EYY (TY + 2)   // 18

typedef float v2f __attribute__((ext_vector_type(2)));
typedef float v8f __attribute__((ext_vector_type(8)));

__device__ __forceinline__ float sigmoidf_(float x) {
  return 1.0f / (1.0f + expf(-x));
}

// ---------------------------------------------------------------------------
// One fused soft-skeleton iteration:
//   eroded = minpool3(x)   (pad +inf)
//   opened = maxpool3(eroded)  (eroded treated as -inf outside the image)
//   x'     = clip(x - (opened - eroded), 0, 1)
// IN_MODE: 0 = raw float, 1 = sigmoid(float logits), 2 = int target -> float
// ---------------------------------------------------------------------------
template <int IN_MODE>
__global__ __launch_bounds__(256) void skel_iter(const void* __restrict__ in_,
                                                 float* __restrict__ out) {
  __shared__ float sx[HYY][HXX];  // x tile with halo 2 (+inf outside image)
  __shared__ float se[EYY][EXX];  // eroded with halo 1 (-inf outside image)

  const int tid = threadIdx.x;
  const int bx  = blockIdx.x * TX;
  const int by  = blockIdx.y * TY;
  const int b   = blockIdx.z;
  const float* inf_ = (const float*)in_;
  const int*   ini_ = (const int*)in_;
  const size_t base = (size_t)b * NPIX;

  // Stage 1: load x (+ mode-specific transform) into LDS, +inf outside image.
  for (int i = tid; i < HXX * HYY; i += 256) {
    const int lx = i % HXX, ly = i / HXX;
    const int gx = bx + lx - 2, gy = by + ly - 2;
    float v = __builtin_inff();
    if (gx >= 0 && gx < WW && gy >= 0 && gy < HH) {
      const size_t idx = base + (size_t)gy * WW + gx;
      if (IN_MODE == 0)      v = inf_[idx];
      else if (IN_MODE == 1) v = sigmoidf_(inf_[idx]);
      else                   v = (float)ini_[idx];
    }
    sx[ly][lx] = v;
  }
  __syncthreads();

  // Stage 2: eroded on (TX+2)x(TY+2); positions outside the image become -inf
  // so the following maxpool ignores them (matches SAME padding semantics).
  for (int i = tid; i < EXX * EYY; i += 256) {
    const int lx = i % EXX, ly = i / EXX;
    const int gx = bx + lx - 1, gy = by + ly - 1;
    float m = -__builtin_inff();
    if (gx >= 0 && gx < WW && gy >= 0 && gy < HH) {
      m =       sx[ly + 0][lx + 0];
      m = fminf(m, sx[ly + 0][lx + 1]);
      m = fminf(m, sx[ly + 0][lx + 2]);
      m = fminf(m, sx[ly + 1][lx + 0]);
      m = fminf(m, sx[ly + 1][lx + 1]);
      m = fminf(m, sx[ly + 1][lx + 2]);
      m = fminf(m, sx[ly + 2][lx + 0]);
      m = fminf(m, sx[ly + 2][lx + 1]);
      m = fminf(m, sx[ly + 2][lx + 2]);
    }
    se[ly][lx] = m;
  }
  __syncthreads();

  // Stage 3: opened + update, write tile.
  for (int i = tid; i < TX * TY; i += 256) {
    const int lx = i % TX, ly = i / TX;
    float o =       se[ly + 0][lx + 0];
    o = fmaxf(o, se[ly + 0][lx + 1]);
    o = fmaxf(o, se[ly + 0][lx + 2]);
    o = fmaxf(o, se[ly + 1][lx + 0]);
    o = fmaxf(o, se[ly + 1][lx + 1]);
    o = fmaxf(o, se[ly + 1][lx + 2]);
    o = fmaxf(o, se[ly + 2][lx + 0]);
    o = fmaxf(o, se[ly + 2][lx + 1]);
    o = fmaxf(o, se[ly + 2][lx + 2]);
    const float er = se[ly + 1][lx + 1];
    const float x  = sx[ly + 2][lx + 2];
    float nx = x - (o - er);
    nx = fminf(fmaxf(nx, 0.0f), 1.0f);
    out[base + (size_t)(by + ly) * WW + (bx + lx)] = nx;
  }
}

// ---------------------------------------------------------------------------
// Deterministic dot-product reduction: part[b*64+c] = sum over a 16K chunk of
// skel * other.  MODE 0: other = (float)int target (tp pass)
//                MODE 1: other = sigmoid(float logits) (ts pass)
// Wave-level reduction uses V_WMMA_F32_16X16X4_F32 against an all-ones B:
// A[m][k] holds lane partials at K0 (lanes 0-15) / K2 (lanes 16-31), zeros
// elsewhere, so D[m][n] = p_m + p_{m+16}; lane 0 holds rows 0-7 of D, lane 16
// rows 8-15; their in-lane 8-sums together give the full wave total.
// EXEC is all-ones here: no divergence before the WMMA.
// ---------------------------------------------------------------------------
template <int MODE>
__global__ __launch_bounds__(256) void reduce_dot(const float* __restrict__ skel,
                                                  const void* __restrict__ other_,
                                                  float* __restrict__ part) {
  __shared__ float lsum[16];  // 8 waves x 2 half-wave sums
  const int    tid  = threadIdx.x;
  const int    c    = blockIdx.x;  // chunk 0..63 within image
  const int    b    = blockIdx.y;  // image 0..15
  const size_t base = (size_t)b * NPIX + (size_t)c * (NPIX / 64);
  const float* of   = (const float*)other_;
  const int*   oi   = (const int*)other_;

  float p = 0.0f;
#pragma unroll 4
  for (int k = 0; k < (NPIX / 64) / 256; ++k) {  // exactly 64 iterations
    const size_t idx = base + (size_t)k * 256 + tid;
    const float  s   = skel[idx];
    const float  o   = (MODE == 0) ? (float)oi[idx] : sigmoidf_(of[idx]);
    p += s * o;
  }

  // WMMA row-sum: all-ones B is layout-agnostic; A carries lane partials.
  v2f a;  a[0] = p;    a[1] = 0.0f;
  v2f bb; bb[0] = 1.0f; bb[1] = 1.0f;
  v8f cacc = {0.0f, 0.0f, 0.0f, 0.0f, 0.0f, 0.0f, 0.0f, 0.0f};
  cacc = __builtin_amdgcn_wmma_f32_16x16x4_f32(
      /*neg_a=*/false, a, /*neg_b=*/false, bb,
      /*c_mod=*/(short)0, cacc, /*reuse_a=*/false, /*reuse_b=*/false);
  const float s8 = cacc[0] + cacc[1] + cacc[2] + cacc[3] +
                   cacc[4] + cacc[5] + cacc[6] + cacc[7];

  const int lane = tid & 31;
  const int wave = tid >> 5;
  if (lane == 0)  lsum[wave * 2 + 0] = s8;  // rows 0-7  -> lanes 0-15 + 16-31 pairs
  if (lane == 16) lsum[wave * 2 + 1] = s8;  // rows 8-15
  __syncthreads();

  if (tid == 0) {
    float t = 0.0f;
    for (int i = 0; i < 16; ++i) t += lsum[i];  // fixed order: deterministic
    part[b * 64 + c] = t;
  }
}

// ---------------------------------------------------------------------------
// Epilogue: fixed-order chunk sums, clDice formula, mean over batch.
// ---------------------------------------------------------------------------
__global__ __launch_bounds__(32) void finalize_cldice(const float* __restrict__ part_tp,
                                                      const float* __restrict__ part_ts,
                                                      float* __restrict__ out) {
  __shared__ float cl[BATCH];
  const int t = threadIdx.x;
  if (t < BATCH) {
    float tp = 0.0f, ts = 0.0f;
    for (int c = 0; c < 64; ++c) {
      tp += part_tp[t * 64 + c];
      ts += part_ts[t * 64 + c];
    }
    const float denom = tp + ts;
    float v = (denom > 0.0f) ? (2.0f * tp * ts + EPSF) / (denom + EPSF) : 1.0f;
    v = fminf(fmaxf(v, 0.0f), 1.0f);
    cl[t] = 1.0f - v;
  }
  __syncthreads();
  if (t == 0) {
    float s = 0.0f;
    for (int i = 0; i < BATCH; ++i) s += cl[i];
    out[0] = s / (float)BATCH;
  }
}

// ---------------------------------------------------------------------------
// Launch: sequential skeletons share two ping-pong buffers (2 x 64 MB, fits
// in the 192 MB L2 so iterations 2..5 are largely cache-resident), then two
// deterministic reductions, then the tiny epilogue.  ws usage: ~134.2 MB.
// ---------------------------------------------------------------------------
extern "C" void kernel_launch(void* const* d_in, const int* in_sizes, int n_in,
                              void* d_out, int out_size, void* d_ws, size_t ws_size,
                              hipStream_t stream) {
  (void)in_sizes; (void)n_in; (void)out_size; (void)ws_size;
  const float* logits = (const float*)d_in[0];
  const int*   target = (const int*)d_in[1];
  float*       out    = (float*)d_out;

  float* ws0     = (float*)d_ws;            // 16M floats
  float* ws1     = ws0 + (size_t)NTOT;      // 16M floats
  float* part_tp = ws1 + (size_t)NTOT;      // 1024 floats
  float* part_ts = part_tp + BATCH * 64;    // 1024 floats

  const dim3 sg(WW / TX, HH / TY, BATCH);   // 16 x 64 x 16 blocks
  const dim3 sb(256);
  const dim3 rg(64, BATCH);

  // skeleton of sigmoid(logits): 5 fused iterations, result lands in ws0
  skel_iter<1><<<sg, sb, 0, stream>>>((const void*)logits, ws0);
  skel_iter<0><<<sg, sb, 0, stream>>>((const void*)ws0, ws1);
  skel_iter<0><<<sg, sb, 0, stream>>>((const void*)ws1, ws0);
  skel_iter<0><<<sg, sb, 0, stream>>>((const void*)ws0, ws1);
  skel_iter<0><<<sg, sb, 0, stream>>>((const void*)ws1, ws0);
  // tp[b] = sum(skel_pred * target)
  reduce_dot<0><<<rg, sb, 0, stream>>>(ws0, (const void*)target, part_tp);

  // skeleton of float(target): reuse the same buffers, result lands in ws0
  skel_iter<2><<<sg, sb, 0, stream>>>((const void*)target, ws0);
  skel_iter<0><<<sg, sb, 0, stream>>>((const void*)ws0, ws1);
  skel_iter<0><<<sg, sb, 0, stream>>>((const void*)ws1, ws0);
  skel_iter<0><<<sg, sb, 0, stream>>>((const void*)ws0, ws1);
  skel_iter<0><<<sg, sb, 0, stream>>>((const void*)ws1, ws0);
  // ts[b] = sum(skel_gt * sigmoid(logits))
  reduce_dot<1><<<rg, sb, 0, stream>>>(ws0, (const void*)logits, part_ts);

  finalize_cldice<<<dim3(1), dim3(32), 0, stream>>>(part_tp, part_ts, out);
}